// GraphConvolution_20306605376076
// MI455X (gfx1250) — compile-verified
//
#include <hip/hip_runtime.h>

typedef __attribute__((ext_vector_type(2))) float v2f;
typedef __attribute__((ext_vector_type(8))) float v8f;

#define N_NODES   100000
#define N_EDGES   1600000
#define IN_FEATS  256
#define OUT_FEATS 128
#define ROW_TILES (N_NODES / 16)   // 6250, exact

// ---------------------------------------------------------------------------
// Kernel 1: out[n][j] = b[j]   (also clears the poisoned output buffer)
// ---------------------------------------------------------------------------
__global__ __launch_bounds__(256) void gc_init_bias(float* __restrict__ out,
                                                    const float* __restrict__ b) {
  int idx = blockIdx.x * 256 + threadIdx.x;
  if (idx < N_NODES * OUT_FEATS) {
    out[idx] = b[idx & (OUT_FEATS - 1)];
  }
}

// ---------------------------------------------------------------------------
// Kernel 2: sup = x @ W  via V_WMMA_F32_16X16X4_F32.
// One wave32 computes a 16x128 output strip (8 accumulator tiles).
// A-matrix layout (16x4 f32): lane<16 -> M=lane, K=k..k+1 ; lane>=16 -> K=k+2..k+3
// B-matrix layout (4x16 f32): v0 = row K (lo lanes)/K+2 (hi lanes), v1 = K+1/K+3
// ---------------------------------------------------------------------------
__global__ __launch_bounds__(256) void gc_gemm_wmma(const float* __restrict__ x,
                                                    const float* __restrict__ w,
                                                    float* __restrict__ sup) {
  const int lane  = threadIdx.x & 31;
  const int wave  = threadIdx.x >> 5;
  const int gwave = blockIdx.x * 8 + wave;
  if (gwave >= ROW_TILES) return;

  const int row_base = gwave * 16;
  const int m  = lane & 15;           // row (A) / col (B,D) within tile
  const int ko = (lane >> 4) << 1;    // 0 for lanes 0-15, 2 for lanes 16-31

  const float* xrow = x + (size_t)(row_base + m) * IN_FEATS;

  v8f acc[8];
#pragma unroll
  for (int t = 0; t < 8; ++t) acc[t] = (v8f){0.f,0.f,0.f,0.f,0.f,0.f,0.f,0.f};

  for (int k = 0; k < IN_FEATS; k += 4) {
    // stream the next chunk of x through L2 (global_prefetch_b8)
    if (k + 32 < IN_FEATS) __builtin_prefetch(xrow + k + 32, 0, 0);

    v2f a = *(const v2f*)(xrow + k + ko);

    const float* wr0 = w + (size_t)(k + ko) * OUT_FEATS + m;  // row k+ko
    const float* wr1 = wr0 + OUT_FEATS;                        // row k+ko+1
#pragma unroll
    for (int t = 0; t < 8; ++t) {
      v2f bm;
      bm.x = wr0[t * 16];
      bm.y = wr1[t * 16];
      acc[t] = __builtin_amdgcn_wmma_f32_16x16x4_f32(
          /*neg_a=*/false, a, /*neg_b=*/false, bm,
          /*c_mod=*/(short)0, acc[t], /*reuse_a=*/false, /*reuse_b=*/false);
    }
  }

  // D layout: VGPR r -> row (row_base + (lane>=16 ? 8:0) + r), col = lane&15
  const int out_row0 = row_base + ((lane >> 4) << 3);
#pragma unroll
  for (int t = 0; t < 8; ++t) {
    float* dst = sup + (size_t)out_row0 * OUT_FEATS + t * 16 + m;
#pragma unroll
    for (int r = 0; r < 8; ++r) {
      dst[(size_t)r * OUT_FEATS] = acc[t][r];
    }
  }
}

// ---------------------------------------------------------------------------
// Kernel 3: COO scatter: out[row[e]] += val[e] * sup[col[e]]
// 64 lanes per edge, float2 per lane -> one fully-coalesced 512B gather/edge.
// ---------------------------------------------------------------------------
__global__ __launch_bounds__(256) void gc_scatter(const float* __restrict__ sup,
                                                  const int* __restrict__ erow,
                                                  const int* __restrict__ ecol,
                                                  const float* __restrict__ eval,
                                                  float* __restrict__ out) {
  int e = blockIdx.x * 4 + (threadIdx.x >> 6);   // 4 edges per 256-thread block
  if (e >= N_EDGES) return;
  const int j = (threadIdx.x & 63) * 2;

  const int   r = erow[e];
  const int   c = ecol[e];
  const float v = eval[e];

  const float* s = sup + (size_t)c * OUT_FEATS + j;
  v2f sv = *(const v2f*)s;

  float* o = out + (size_t)r * OUT_FEATS + j;
  atomicAdd(o,     v * sv.x);
  atomicAdd(o + 1, v * sv.y);
}

// ---------------------------------------------------------------------------
extern "C" void kernel_launch(void* const* d_in, const int* in_sizes, int n_in,
                              void* d_out, int out_size, void* d_ws, size_t ws_size,
                              hipStream_t stream) {
  const float* x    = (const float*)d_in[0];
  const int*   erow = (const int*)  d_in[1];
  const int*   ecol = (const int*)  d_in[2];
  const float* eval = (const float*)d_in[3];
  const float* w    = (const float*)d_in[4];
  const float* b    = (const float*)d_in[5];

  float* out = (float*)d_out;
  float* sup = (float*)d_ws;   // 100000*128*4 = 51.2 MB scratch

  gc_init_bias<<<(N_NODES * OUT_FEATS + 255) / 256, 256, 0, stream>>>(out, b);
  gc_gemm_wmma<<<(ROW_TILES + 7) / 8, 256, 0, stream>>>(x, w, sup);
  gc_scatter<<<(N_EDGES + 3) / 4, 256, 0, stream>>>(sup, erow, ecol, eval, out);
}